// MemoryAugmentedLayer_87677462380893
// MI455X (gfx1250) — compile-verified
//
#include <hip/hip_runtime.h>
#include <math.h>

#define BB 8
#define SS 32
#define DD 256
#define MM 2048
#define TOPK 8

typedef __attribute__((ext_vector_type(2))) float v2f;
typedef __attribute__((ext_vector_type(8))) float v8f;

// ---------------------------------------------------------------------------
// Kernel 1: replicate initial memory [M,D] into per-batch state [B,M,D]
// ---------------------------------------------------------------------------
__global__ __launch_bounds__(256) void mal_init_mem(const float* __restrict__ memory,
                                                    float* __restrict__ mem) {
  const size_t n = (size_t)MM * DD / 4;  // float4 elements per batch
  size_t i = (size_t)blockIdx.x * blockDim.x + threadIdx.x;
  if (i < n) {
    const float4* src = (const float4*)memory;
    float4* dst = (float4*)mem;
    float4 val = src[i];
#pragma unroll
    for (int b = 0; b < BB; ++b) dst[(size_t)b * n + i] = val;
  }
}

// ---------------------------------------------------------------------------
// Kernel 2: one-time projection of the (batch-shared) initial memory:
//   k0 = memory @ Wk + bk ; v0 = memory @ Wv + bv   (WMMA fp32 16x16x4)
// grid = M/16 blocks, 256 threads (8 waves); wave w covers col-tiles {w, w+8}.
// ---------------------------------------------------------------------------
__global__ __launch_bounds__(256) void mal_proj_kv(
    const float* __restrict__ memory,
    const float* __restrict__ Wk, const float* __restrict__ bk,
    const float* __restrict__ Wv, const float* __restrict__ bv,
    float* __restrict__ kbuf, float* __restrict__ vbuf) {
  const int m0   = blockIdx.x * 16;
  const int wave = threadIdx.x >> 5;
  const int lane = threadIdx.x & 31;
  const int hl   = lane & 15;
  const bool hi  = lane >= 16;

  for (int sel = 0; sel < 2; ++sel) {
    const float* W    = sel ? Wv : Wk;
    const float* bias = sel ? bv : bk;
    float* dst        = sel ? vbuf : kbuf;
    for (int ct = wave; ct < 16; ct += 8) {
      const int c0 = ct * 16;
      v8f acc = {};
      const float* arow = memory + (size_t)(m0 + hl) * DD;
      for (int kk = 0; kk < DD; kk += 4) {
        const int ka = kk + (hi ? 2 : 0);
        v2f a;  a.x = arow[ka];                       a.y = arow[ka + 1];
        v2f bfr; bfr.x = W[(size_t)(ka    ) * DD + c0 + hl];
                 bfr.y = W[(size_t)(ka + 1) * DD + c0 + hl];
        acc = __builtin_amdgcn_wmma_f32_16x16x4_f32(false, a, false, bfr,
                                                    (short)0, acc, false, false);
      }
      const float bval  = bias[c0 + hl];
      const int   rbase = hi ? 8 : 0;  // D layout: VGPR r = row r (lo lanes) / r+8 (hi lanes)
#pragma unroll
      for (int r = 0; r < 8; ++r) {
        const int row = m0 + rbase + r;
        const float val = acc[r] + bval;
#pragma unroll
        for (int b = 0; b < BB; ++b)
          dst[((size_t)b * MM + row) * DD + c0 + hl] = val;
      }
    }
  }
}

// ---------------------------------------------------------------------------
// Kernel 3: the whole sequential scan. One block (1024 thr = 32 waves) per
// batch; batches are independent so block barriers suffice.
// ---------------------------------------------------------------------------
__global__ __launch_bounds__(1024) void mal_scan(
    const float* __restrict__ x,
    const float* __restrict__ Wq, const float* __restrict__ bq,
    const float* __restrict__ Wk, const float* __restrict__ bk,
    const float* __restrict__ Wv, const float* __restrict__ bv,
    const float* __restrict__ Wg, const float* __restrict__ bg,
    float* __restrict__ mem, float* __restrict__ kbuf, float* __restrict__ vbuf,
    float* __restrict__ out) {
  const int b   = blockIdx.x;
  const int tid = threadIdx.x;
  float* memb = mem + (size_t)b * MM * DD;

  __shared__ __align__(16) float xs[DD];
  __shared__ __align__(16) float qs[DD];
  __shared__ __align__(16) float os[DD];
  __shared__ __align__(16) float gs[DD];
  __shared__ __align__(16) float attn[MM];
  __shared__ __align__(16) float part[16][DD];   // partial sums (q/gate use 4 rows, out uses 16)
  __shared__ float redv[1024];
  __shared__ int   redi[1024];
  __shared__ __align__(16) float news[TOPK][DD];
  __shared__ int   topidx[TOPK];
  __shared__ float smax, ssum;

  for (int t = 0; t < SS; ++t) {
    // ---- load x_t ----
    if (tid < DD) xs[tid] = x[((size_t)b * SS + t) * DD + tid];
    __syncthreads();

    // ---- q = x_t @ Wq + bq (4 K-chunks x 256 outputs) ----
    {
      const int d = tid & (DD - 1), chunk = tid >> 8;
      float acc = 0.f;
      for (int i = chunk * 64; i < chunk * 64 + 64; ++i) acc += xs[i] * Wq[(size_t)i * DD + d];
      part[chunk][d] = acc;
      __syncthreads();
      if (tid < DD) qs[tid] = part[0][tid] + part[1][tid] + part[2][tid] + part[3][tid] + bq[tid];
      __syncthreads();
    }

    // ---- logits[m] = <q, k[b,m]> / 16   (float4 row dots, B128 loads) ----
    for (int m = tid; m < MM; m += 1024) {
      const float4* kr = (const float4*)(kbuf + ((size_t)b * MM + m) * DD);
      __builtin_prefetch(kr + DD / 4, 0, 0);
      const float4* q4 = (const float4*)qs;
      float4 a4 = make_float4(0.f, 0.f, 0.f, 0.f);
      for (int i = 0; i < DD / 4; ++i) {
        const float4 kv = kr[i];
        const float4 qv = q4[i];
        a4.x += qv.x * kv.x; a4.y += qv.y * kv.y;
        a4.z += qv.z * kv.z; a4.w += qv.w * kv.w;
      }
      attn[m] = (a4.x + a4.y + a4.z + a4.w) * 0.0625f;  // 1/sqrt(256)
    }
    __syncthreads();

    // ---- softmax over M=2048 ----
    redv[tid] = fmaxf(attn[tid], attn[tid + 1024]);
    __syncthreads();
    for (int s = 512; s > 0; s >>= 1) {
      if (tid < s) redv[tid] = fmaxf(redv[tid], redv[tid + s]);
      __syncthreads();
    }
    if (tid == 0) smax = redv[0];
    __syncthreads();
    {
      const float e0 = __expf(attn[tid] - smax);
      const float e1 = __expf(attn[tid + 1024] - smax);
      attn[tid] = e0; attn[tid + 1024] = e1;
      redv[tid] = e0 + e1;
    }
    __syncthreads();
    for (int s = 512; s > 0; s >>= 1) {
      if (tid < s) redv[tid] += redv[tid + s];
      __syncthreads();
    }
    if (tid == 0) ssum = redv[0];
    __syncthreads();
    {
      const float inv = 1.f / ssum;
      attn[tid] *= inv; attn[tid + 1024] *= inv;
    }
    __syncthreads();

    // ---- out[d] = sum_m attn[m]*v[b,m,d]  (16 m-chunks x 64 col-groups, B128) ----
    {
      const int dg = (tid & 63) * 4;   // columns d..d+3
      const int chunk = tid >> 6;      // 0..15, 128 m's each
      float4 acc = make_float4(0.f, 0.f, 0.f, 0.f);
      const float* vb = vbuf + (size_t)b * MM * DD;
      for (int m = chunk * 128; m < chunk * 128 + 128; ++m) {
        const float a = attn[m];
        const float4 vv = *(const float4*)(vb + (size_t)m * DD + dg);
        acc.x += a * vv.x; acc.y += a * vv.y; acc.z += a * vv.z; acc.w += a * vv.w;
      }
      part[chunk][dg + 0] = acc.x; part[chunk][dg + 1] = acc.y;
      part[chunk][dg + 2] = acc.z; part[chunk][dg + 3] = acc.w;
      __syncthreads();
      if (tid < DD) {
        float o = 0.f;
#pragma unroll
        for (int c = 0; c < 16; ++c) o += part[c][tid];
        os[tid] = o;
        out[((size_t)b * SS + t) * DD + tid] = o;
      }
      __syncthreads();
    }

    // ---- gate g = sigmoid(cat(x_t,out) @ Wg + bg) ----
    {
      const int d = tid & (DD - 1), chunk = tid >> 8;
      float acc = 0.f;
      for (int i = chunk * 128; i < chunk * 128 + 128; ++i) {
        const float xi = (i < DD) ? xs[i] : os[i - DD];
        acc += xi * Wg[(size_t)i * DD + d];
      }
      part[chunk][d] = acc;
      __syncthreads();
      if (tid < DD) {
        const float z = part[0][tid] + part[1][tid] + part[2][tid] + part[3][tid] + bg[tid];
        gs[tid] = 1.f / (1.f + __expf(-z));
      }
      __syncthreads();
    }

    // ---- top-8 (destroys attn; no longer needed) ----
    for (int j = 0; j < TOPK; ++j) {
      float v0 = attn[tid], v1 = attn[tid + 1024];
      float bv_; int bi;
      if (v0 >= v1) { bv_ = v0; bi = tid; } else { bv_ = v1; bi = tid + 1024; }
      redv[tid] = bv_; redi[tid] = bi;
      __syncthreads();
      for (int s = 512; s > 0; s >>= 1) {
        if (tid < s) {
          const float ov = redv[tid + s]; const int oi = redi[tid + s];
          if (ov > redv[tid] || (ov == redv[tid] && oi < redi[tid])) { redv[tid] = ov; redi[tid] = oi; }
        }
        __syncthreads();
      }
      if (tid == 0) { topidx[j] = redi[0]; attn[redi[0]] = -1.f; }
      __syncthreads();
    }

    // ---- memory row update: new = (1-g)*old + g*x_t  (float4) ----
    if (tid < TOPK * DD / 4) {
      const int j = tid >> 6, d4 = (tid & 63) * 4;
      const int row = topidx[j];
      float* mr = memb + (size_t)row * DD + d4;
      const float4 old = *(const float4*)mr;
      const float4 xv = *(const float4*)(xs + d4);
      const float4 gv = *(const float4*)(gs + d4);
      float4 nv;
      nv.x = (1.f - gv.x) * old.x + gv.x * xv.x;
      nv.y = (1.f - gv.y) * old.y + gv.y * xv.y;
      nv.z = (1.f - gv.z) * old.z + gv.z * xv.z;
      nv.w = (1.f - gv.w) * old.w + gv.w * xv.w;
      *(float4*)(&news[j][d4]) = nv;
      *(float4*)mr = nv;
    }
    __syncthreads();

    // ---- incremental k/v re-projection of the 8 new rows (WMMA fp32) ----
    {
      const int wave = tid >> 5, lane = tid & 31;
      const int hl = lane & 15;  const bool hi = lane >= 16;
      const float* W    = (wave < 16) ? Wk : Wv;
      const float* bias = (wave < 16) ? bk : bv;
      float* dst        = (wave < 16) ? kbuf : vbuf;
      const int c0 = (wave & 15) * 16;
      v8f acc = {};
      for (int kk = 0; kk < DD; kk += 4) {
        const int ka = kk + (hi ? 2 : 0);
        v2f a;
        if (hl < TOPK) { a.x = news[hl][ka]; a.y = news[hl][ka + 1]; }
        else           { a.x = 0.f;          a.y = 0.f; }
        v2f bfr; bfr.x = W[(size_t)(ka    ) * DD + c0 + hl];
                 bfr.y = W[(size_t)(ka + 1) * DD + c0 + hl];
        acc = __builtin_amdgcn_wmma_f32_16x16x4_f32(false, a, false, bfr,
                                                    (short)0, acc, false, false);
      }
      if (!hi) {  // rows 0..7 of D live in VGPRs 0..7 on the low half-wave
        const float bval = bias[c0 + hl];
#pragma unroll
        for (int r = 0; r < TOPK; ++r) {
          const int row = topidx[r];
          dst[((size_t)b * MM + row) * DD + c0 + hl] = acc[r] + bval;
        }
      }
    }
    __syncthreads();
  }
}

// ---------------------------------------------------------------------------
extern "C" void kernel_launch(void* const* d_in, const int* in_sizes, int n_in,
                              void* d_out, int out_size, void* d_ws, size_t ws_size,
                              hipStream_t stream) {
  const float* x      = (const float*)d_in[0];
  const float* memory = (const float*)d_in[1];
  const float* Wq = (const float*)d_in[2]; const float* bq = (const float*)d_in[3];
  const float* Wk = (const float*)d_in[4]; const float* bk = (const float*)d_in[5];
  const float* Wv = (const float*)d_in[6]; const float* bv = (const float*)d_in[7];
  const float* Wg = (const float*)d_in[8]; const float* bg = (const float*)d_in[9];
  float* out = (float*)d_out;

  char* ws = (char*)d_ws;
  const size_t sz = (size_t)BB * MM * DD * sizeof(float);  // 16.8 MB each
  float* mem  = (float*)(ws);
  float* kbuf = (float*)(ws + sz);
  float* vbuf = (float*)(ws + 2 * sz);

  const int initBlocks = (int)(((size_t)MM * DD / 4 + 255) / 256);
  mal_init_mem<<<initBlocks, 256, 0, stream>>>(memory, mem);
  mal_proj_kv<<<MM / 16, 256, 0, stream>>>(memory, Wk, bk, Wv, bv, kbuf, vbuf);
  mal_scan<<<BB, 1024, 0, stream>>>(x, Wq, bq, Wk, bk, Wv, bv, Wg, bg,
                                    mem, kbuf, vbuf, out);
}